// ForceGraphAttentionLayer_412316860914
// MI455X (gfx1250) — compile-verified
//
#include <hip/hip_runtime.h>

// MI455X / gfx1250, wave32.
//
// Reference: ForceGraphAttentionLayer  (B=8, N=1024, N_IN=64)
//   h  = x @ [a[:64] | a[64:128]]          -> [B,N,16]   (WMMA f32 16x16x4)
//   pre= hi + hj + adj*a[128]; att = lrelu(pre)@a2
//   out[b,i,:] = sum_j att[b,i,j] * normalize(c_i - c_j)
//
// Roofline: adj (32 MB) is the only real HBM stream (~1.4 us at 23.3 TB/s).
// Per-batch hj/c working set (44 KB) staged in LDS per block, TRANSPOSED
// ([k][j] / [d][j]) so the hot loop reads aligned ds_load_b128 with no bank
// conflicts. Each of the 256 threads owns one float4 of the adj row -> one
// coalesced global_load_b128 per thread, j-loop fully unrolled.

typedef float v2f __attribute__((ext_vector_type(2)));
typedef float v8f __attribute__((ext_vector_type(8)));

#define B_   8
#define N_   1024
#define NIN  64

// ---------------------------------------------------------------------------
// Kernel 1: h[r, 0:8]  = x[r,:] @ a[0:64]     (hi)
//           h[r, 8:16] = x[r,:] @ a[64:128]   (hj)
// One wave32 per 16-row tile of the flattened [B*N, 64] x matrix.
// K=64 handled as 16 chained V_WMMA_F32_16X16X4_F32 ops (exact fp32).
// ---------------------------------------------------------------------------
__global__ __launch_bounds__(32) void gat_h_wmma(const float* __restrict__ x,
                                                 const float* __restrict__ a,
                                                 float* __restrict__ h) {
  const int lane = threadIdx.x;       // 0..31
  const int half = lane >> 4;         // 0 | 1
  const int lr   = lane & 15;
  const int row0 = blockIdx.x * 16;   // first of 16 rows of x handled here
  const float* xr = x + (size_t)(row0 + lr) * NIN;

  // Fused weight W[f][n] (64x16): n<8 -> a[f,n] ; n>=8 -> a[64+f, n-8]
  const int ca = lr & 7;

  v8f acc = {};
#pragma unroll
  for (int it = 0; it < 16; ++it) {
    // A 16x4 f32 layout: lanes 0-15 hold row M=lane, K = {it*4+0, it*4+1};
    //                    lanes 16-31 hold row M=lane-16, K = {it*4+2, it*4+3}.
    const int k0 = it * 4 + half * 2;
    v2f af, bf;
    af.x = xr[k0];
    af.y = xr[k0 + 1];
    // B 4x16 f32 layout mirrors A over N: lane group selects K pair, lr = col.
    const int fa = (lr < 8) ? k0 : (NIN + k0);
    bf.x = a[fa * 8 + ca];
    bf.y = a[(fa + 1) * 8 + ca];
    // 8-arg form: (neg_a, A, neg_b, B, c_mod, C, reuse_a, reuse_b)
    acc = __builtin_amdgcn_wmma_f32_16x16x4_f32(false, af, false, bf,
                                                (short)0, acc, false, false);
  }

  // D 16x16 f32: lane group 0 -> M = v, group 1 -> M = 8+v; N = lr.
  float* hrow = h + (size_t)row0 * 16;
#pragma unroll
  for (int v = 0; v < 8; ++v) {
    hrow[(size_t)(half * 8 + v) * 16 + lr] = acc[v];
  }
}

// ---------------------------------------------------------------------------
// Kernel 2: one 256-thread block per (b, i). hj/c staged transposed in LDS,
// adj row consumed as one float4 per thread, wave32 shfl_xor reduction.
// ---------------------------------------------------------------------------
__global__ __launch_bounds__(256) void gat_att_force(
    const float* __restrict__ adj, const float* __restrict__ c,
    const float* __restrict__ h, const float* __restrict__ a,
    const float* __restrict__ a2, float* __restrict__ out) {
  __shared__ float sh_hj[8 * N_];   // 32 KB, transposed: [k][j]
  __shared__ float sh_c[3 * N_];    // 12 KB, transposed: [d][j]
  __shared__ float sh_red[8 * 3];   // one partial per wave

  const int tid = threadIdx.x;
  const int bi  = blockIdx.x;       // b*N + i
  const int b   = bi >> 10;
  const int i   = bi & (N_ - 1);
  const int j0  = tid * 4;          // this thread's 4 columns

  // Kick off the adj stream first: its latency overlaps the LDS staging.
  const float4 av4 = *(const float4*)(adj + (size_t)bi * N_ + j0);

  // Stage hj transposed: coalesced b128 global reads, scattered b32 LDS stores.
  const float* hb = h + (size_t)b * N_ * 16;
  for (int idx = tid; idx < N_ * 2; idx += 256) {
    const int j  = idx >> 1;
    const int q4 = (idx & 1) * 4;                       // k-quad 0..3 or 4..7
    const float4 v = *(const float4*)(hb + (size_t)j * 16 + 8 + q4);
    sh_hj[(q4 + 0) * N_ + j] = v.x;
    sh_hj[(q4 + 1) * N_ + j] = v.y;
    sh_hj[(q4 + 2) * N_ + j] = v.z;
    sh_hj[(q4 + 3) * N_ + j] = v.w;
  }
  // Stage c transposed.
  const float* cb = c + (size_t)b * N_ * 3;
  for (int j = tid; j < N_; j += 256) {
    sh_c[0 * N_ + j] = cb[j * 3 + 0];
    sh_c[1 * N_ + j] = cb[j * 3 + 1];
    sh_c[2 * N_ + j] = cb[j * 3 + 2];
  }

  float hik[8], aak[8], a2k[8];
#pragma unroll
  for (int k = 0; k < 8; ++k) {
    hik[k] = hb[(size_t)i * 16 + k];   // hi contribution for row i
    aak[k] = a[128 * 8 + k];           // a[2*n_in] row (adj weight)
    a2k[k] = a2[k];
  }
  const float cix = cb[i * 3 + 0];
  const float ciy = cb[i * 3 + 1];
  const float ciz = cb[i * 3 + 2];

  __syncthreads();

  // Hot block: 4 j's per thread, all LDS traffic as aligned ds_load_b128.
  float av[4], hj[8][4], cjx[4], cjy[4], cjz[4];
  *(float4*)av = av4;
#pragma unroll
  for (int k = 0; k < 8; ++k)
    *(float4*)hj[k] = *(const float4*)(sh_hj + k * N_ + j0);
  *(float4*)cjx = *(const float4*)(sh_c + 0 * N_ + j0);
  *(float4*)cjy = *(const float4*)(sh_c + 1 * N_ + j0);
  *(float4*)cjz = *(const float4*)(sh_c + 2 * N_ + j0);

  float ox = 0.f, oy = 0.f, oz = 0.f;
#pragma unroll
  for (int q = 0; q < 4; ++q) {
    float att = 0.f;
#pragma unroll
    for (int k = 0; k < 8; ++k) {
      const float pre = hik[k] + hj[k][q] + av[q] * aak[k];
      const float lrl = fmaxf(pre, 0.01f * pre);  // leaky_relu(pre, 0.01)
      att += lrl * a2k[k];
    }
    const float dx = cix - cjx[q];
    const float dy = ciy - cjy[q];
    const float dz = ciz - cjz[q];
    const float nrm = sqrtf(dx * dx + dy * dy + dz * dz);
    const float w = att / (nrm + 1e-8f);  // j==i: dx=dy=dz=0 -> contributes 0
    ox += w * dx;
    oy += w * dy;
    oz += w * dz;
  }

  // Wave32 butterfly reduction (lane-permute HW, no LDS), then 8-wave combine.
#pragma unroll
  for (int off = 16; off > 0; off >>= 1) {
    ox += __shfl_xor(ox, off, 32);
    oy += __shfl_xor(oy, off, 32);
    oz += __shfl_xor(oz, off, 32);
  }
  const int wave = tid >> 5;
  if ((tid & 31) == 0) {
    sh_red[wave * 3 + 0] = ox;
    sh_red[wave * 3 + 1] = oy;
    sh_red[wave * 3 + 2] = oz;
  }
  __syncthreads();
  if (tid == 0) {
    float sx = 0.f, sy = 0.f, sz = 0.f;
#pragma unroll
    for (int w8 = 0; w8 < 8; ++w8) {
      sx += sh_red[w8 * 3 + 0];
      sy += sh_red[w8 * 3 + 1];
      sz += sh_red[w8 * 3 + 2];
    }
    out[(size_t)bi * 3 + 0] = sx;
    out[(size_t)bi * 3 + 1] = sy;
    out[(size_t)bi * 3 + 2] = sz;
  }
}

// ---------------------------------------------------------------------------
extern "C" void kernel_launch(void* const* d_in, const int* in_sizes, int n_in,
                              void* d_out, int out_size, void* d_ws,
                              size_t ws_size, hipStream_t stream) {
  (void)in_sizes; (void)n_in; (void)out_size; (void)ws_size;
  const float* adj = (const float*)d_in[0];  // [B,N,N]
  const float* c   = (const float*)d_in[1];  // [B,N,3]
  const float* x   = (const float*)d_in[2];  // [B,N,64]
  const float* a   = (const float*)d_in[3];  // [129,8]
  const float* a2  = (const float*)d_in[4];  // [8]
  float* out = (float*)d_out;                // [B,N,3]
  float* h = (float*)d_ws;                   // [B,N,16] = 512 KB scratch

  gat_h_wmma<<<(B_ * N_) / 16, 32, 0, stream>>>(x, a, h);
  gat_att_force<<<B_ * N_, 256, 0, stream>>>(adj, c, h, a, a2, out);
}